// MeanshiftCluster_2_62234076119455
// MI455X (gfx1250) — compile-verified
//
#include <hip/hip_runtime.h>

// MeanshiftCluster on gfx1250 (MI455X).
// 3 bilateral-filter iterations, 19x19 window, replication padding.
// Design: 16x16 output tile per 256-thread block; 34x34 halo staged in LDS via
// gfx1250 async global->LDS loads (ASYNCcnt path); 361-entry spatial-Gaussian
// LUT built once per block in LDS; one v_exp_f32 per tap for the range kernel.

#define HH    192
#define WW    192
#define IMGS  2
#define RAD   9
#define DIAM  19               // 2*RAD+1
#define TILE  16
#define HREG  (TILE + 2*RAD)   // 34
#define NHALO (HREG*HREG)      // 1156
#define NTAPS (DIAM*DIAM)      // 361

// exp(-2*d^2)            = exp2(-2*log2(e) * d^2)
#define RNG_COEF  (-2.8853900818f)
#define RNG_AMP   (0.79788456080f)   // 1/(0.5*sqrt(2*pi))
#define RNG_THR   (0.48394144900f)   // exp(-0.5)*RNG_AMP
// spatial: sigma_s^2 = 2*81/1.5^2 = 72 exactly
// exp(-r2/144)           = exp2(-r2 * log2(e)/144)
#define SPT_COEF  (-0.01001871556f)
#define SPT_AMP   (0.04701582000f)   // 1/(sqrt(72)*sqrt(2*pi))

#if defined(__has_builtin)
#if __has_builtin(__builtin_amdgcn_global_load_async_to_lds_b32)
#define HAVE_ASYNC_BUILTIN 1
#endif
#endif

typedef __attribute__((address_space(1))) int gbl_i32;   // prints as "__device__ int *"
typedef __attribute__((address_space(3))) int lds_i32;   // LDS int *

__device__ __forceinline__ void async_f32_to_lds(const float* gp, float* lp) {
    // low 32 bits of a flat shared pointer == LDS byte offset (aperture form)
    unsigned ldsoff = (unsigned)(unsigned long long)(const void*)lp;
#if defined(HAVE_ASYNC_BUILTIN)
    __builtin_amdgcn_global_load_async_to_lds_b32(
        (gbl_i32*)(unsigned long long)(const void*)gp,
        (lds_i32*)ldsoff, 0, 0);
#else
    asm volatile("global_load_async_to_lds_b32 %0, %1, off"
                 :
                 : "v"(ldsoff), "v"((unsigned long long)(const void*)gp)
                 : "memory");
#endif
}

__device__ __forceinline__ void wait_async_all() {
    asm volatile("s_wait_asynccnt 0" ::: "memory");
}

__global__ __launch_bounds__(256)
void ms_iter_kernel(const float* __restrict__ src, float* __restrict__ dst) {
    __shared__ float tile[NHALO];   // 34x34 halo region
    __shared__ float sptw[NTAPS];   // 19x19 spatial Gaussian LUT

    const int lx  = threadIdx.x;            // 0..15
    const int ly  = threadIdx.y;            // 0..15
    const int tid = ly * TILE + lx;         // 0..255
    const int bx  = blockIdx.x * TILE;
    const int by  = blockIdx.y * TILE;
    const int img = blockIdx.z;
    const float* s = src + img * (HH * WW);

    // --- spatial weight LUT (105 threads do 2 entries, the other 151 do 1) ---
    for (int i = tid; i < NTAPS; i += 256) {
        int dy = i / DIAM - RAD;
        int dx = i % DIAM - RAD;
        float r2 = (float)(dx * dx + dy * dy);
        sptw[i] = SPT_AMP * __builtin_amdgcn_exp2f(SPT_COEF * r2);
    }

    // --- halo load with edge replication, async global->LDS (ASYNCcnt) ---
    for (int i = tid; i < NHALO; i += 256) {
        int ry = i / HREG;
        int rx = i - ry * HREG;
        int gy = min(max(by - RAD + ry, 0), HH - 1);
        int gx = min(max(bx - RAD + rx, 0), WW - 1);
        async_f32_to_lds(s + gy * WW + gx, &tile[i]);
    }
    wait_async_all();      // this wave's async LDS writes have landed
    __syncthreads();       // all waves' writes visible

    // --- 361-tap weighted mean; one v_exp_f32 per tap ---
    const float c = tile[(ly + RAD) * HREG + (lx + RAD)];
    float num = 0.0f, den = 0.0f;
#pragma unroll 1
    for (int dy = 0; dy < DIAM; ++dy) {
        const float* row = &tile[(ly + dy) * HREG + lx];
        const float* sw  = &sptw[dy * DIAM];
#pragma unroll
        for (int dx = 0; dx < DIAM; ++dx) {
            float n  = row[dx];
            float d  = c - n;
            float e  = __builtin_amdgcn_exp2f(RNG_COEF * d * d);
            float rw = __builtin_fmaf(RNG_AMP, e, -RNG_THR);
            rw       = fmaxf(rw, 0.0f);
            float w  = rw * sw[dx];
            num      = __builtin_fmaf(w, n, num);
            den     += w;
        }
    }
    dst[img * HH * WW + (by + ly) * WW + (bx + lx)] = num / (den + 1e-8f);
}

extern "C" void kernel_launch(void* const* d_in, const int* in_sizes, int n_in,
                              void* d_out, int out_size, void* d_ws, size_t ws_size,
                              hipStream_t stream) {
    (void)in_sizes; (void)n_in; (void)out_size; (void)ws_size;
    const float* src = (const float*)d_in[0];   // [2,1,192,192] fp32
    float* out = (float*)d_out;                 // [2,1,192,192] fp32
    float* tmp = (float*)d_ws;                  // needs 2*192*192*4 = 294912 B

    dim3 grid(WW / TILE, HH / TILE, IMGS);      // 12 x 12 x 2
    dim3 block(TILE, TILE, 1);                  // 256 threads = 8 wave32

    // iter 1: src -> out ; iter 2: out -> ws ; iter 3: ws -> out
    ms_iter_kernel<<<grid, block, 0, stream>>>(src, out);
    ms_iter_kernel<<<grid, block, 0, stream>>>(out, tmp);
    ms_iter_kernel<<<grid, block, 0, stream>>>(tmp, out);
}